// CosSim2D_17575006175389
// MI455X (gfx1250) — compile-verified
//
#include <hip/hip_runtime.h>

// CosSim2D (B=8, H=W=224, C=32, F=32, K=3, same-pad) on gfx1250.
// GEMM: M=401408, K=288, N=32, fp32 via V_WMMA_F32_16X16X4_F32 (wave32).
// Roofline: ~7.4 GFLOP vs ~103 MB HBM traffic -> a few us either way; fp32
// WMMA keeps reference precision while still beating the memory floor by <4x.

typedef __attribute__((ext_vector_type(2))) float v2f;
typedef __attribute__((ext_vector_type(8))) float v8f;

#define BB 8
#define HH 224
#define WW 224
#define CC 32
#define FF 32
#define KDIM 288      // C * 3 * 3 (kernel-position-major, channel-minor)
#define KSTEPS 72     // KDIM / 4
#define TILE 64       // pixels (along W) per block
#define NTHREADS 128  // 4 wave32s; each wave -> 16 pixels x 32 filters
#define TCOLS 66      // TILE + 2 halo columns

// d_ws float layout
#define WS_WNP 0      // 9216 floats: normalized W in K-pair layout [k/2][f][k&1]
#define WS_PEXP 9216  // 32 floats: p[f]^2 / 100
#define WS_QT 9248    // 1 float:  q^2 / 10

__global__ __launch_bounds__(32) void cos_prep_w(const float* __restrict__ w,
                                                 const float* __restrict__ p,
                                                 const float* __restrict__ q,
                                                 float* __restrict__ ws) {
  const int f = threadIdx.x;  // 0..31
  const float qv = q[0];
  const float qt = qv * qv * 0.1f;
  float s = 0.f;
  for (int k = 0; k < KDIM; ++k) {
    const float t = w[k * FF + f];
    s = fmaf(t, t, s);
  }
  const float inv = 1.f / (sqrtf(fmaxf(s, 1e-12f)) + qt);
  // Pair layout matches assumed WMMA B operand: B.x = wn[2*kk2][n], B.y = wn[2*kk2+1][n]
  for (int k = 0; k < KDIM; ++k) {
    ws[WS_WNP + (k >> 1) * (2 * FF) + f * 2 + (k & 1)] = w[k * FF + f] * inv;
  }
  const float pv = p[f];
  ws[WS_PEXP + f] = pv * pv * 0.01f;
  if (f == 0) ws[WS_QT] = qt;
}

__device__ __forceinline__ float signpow(float v, float e) {
  const float a = fabsf(v) + 1e-12f;
  // v_log_f32 / v_exp_f32 are base-2 hardware transcendentals on AMD.
  const float r = __builtin_amdgcn_exp2f(e * __builtin_amdgcn_logf(a));
  return v > 0.f ? r : (v < 0.f ? -r : 0.f);
}

__global__ __launch_bounds__(NTHREADS) void cos_sim_main(const float* __restrict__ img,
                                                         const float* __restrict__ ws,
                                                         float* __restrict__ out) {
  __shared__ float imgT[3 * TCOLS * CC];  // 6336 f: rows h-1..h+1, cols w0-1..w0+64, 32 ch
  __shared__ float wnp[KDIM * FF];        // 9216 f: normalized weights, K-pair layout
  __shared__ float rxn[TILE];             // 64 f: 1/(||patch|| + q^2/10) per pixel
  __shared__ float pexp[FF];              // 32 f
  // total 62592 B static LDS (< 64KB; WGP has 320KB)

  const int tid = threadIdx.x;
  const int w0 = blockIdx.x * TILE;
  const int h = blockIdx.y;
  const int b = blockIdx.z;

  // ---- stage normalized weights + exponents into LDS (float4 copies) ----
  {
    const float4* src = (const float4*)(ws + WS_WNP);
    float4* dst = (float4*)wnp;
    for (int i = tid; i < (KDIM * FF) / 4; i += NTHREADS) dst[i] = src[i];
    if (tid < FF) pexp[tid] = ws[WS_PEXP + tid];
  }
  // ---- stage image tile with zero halo: 3 rows x 66 cols x 8 float4 ----
  {
    float4* dst = (float4*)imgT;
    for (int i = tid; i < 3 * TCOLS * 8; i += NTHREADS) {
      const int c4 = i & 7;
      const int col = (i >> 3) % TCOLS;
      const int row = (i >> 3) / TCOLS;
      const int gh = h + row - 1;
      const int gw = w0 + col - 1;
      float4 v = make_float4(0.f, 0.f, 0.f, 0.f);
      if (gh >= 0 && gh < HH && gw >= 0 && gw < WW) {
        const float4* g = (const float4*)(img + (((size_t)b * HH + gh) * WW + gw) * CC);
        v = g[c4];
      }
      dst[(row * TCOLS + col) * 8 + c4] = v;
    }
  }
  __syncthreads();

  // ---- per-pixel reciprocal patch norm from the LDS tile ----
  if (tid < TILE) {
    float s = 0.f;
    for (int r = 0; r < 3; ++r) {
      for (int dx = 0; dx < 3; ++dx) {
        const float* pp = &imgT[(r * TCOLS + tid + dx) * CC];
#pragma unroll
        for (int c = 0; c < CC; ++c) s = fmaf(pp[c], pp[c], s);
      }
    }
    rxn[tid] = 1.f / (sqrtf(fmaxf(s, 1e-12f)) + ws[WS_QT]);
  }
  __syncthreads();

  // ---- WMMA GEMM: each wave does 16 pixels x 32 filters, K=288 in steps of 4 ----
  const int wave = tid >> 5;
  const int lane = tid & 31;
  const int wbase = w0 + wave * 16;
  if (wbase < WW) {  // wave-uniform: EXEC stays all-ones around WMMA
    const int mi = lane & 15;      // M row (pixel within wave tile)
    const int khalf = lane >> 4;   // 0: K+0/K+1, 1: K+2/K+3 (A & B layouts)
    const int nidx = lane & 15;    // N column within 16-wide tile
    const float rn = rxn[wave * 16 + mi];  // fold 1/x_norm into A rows
    const int coli = wave * 16 + mi;       // LDS tile column for dx=0

    v8f acc0 = {0.f, 0.f, 0.f, 0.f, 0.f, 0.f, 0.f, 0.f};
    v8f acc1 = {0.f, 0.f, 0.f, 0.f, 0.f, 0.f, 0.f, 0.f};

#pragma unroll
    for (int kk = 0; kk < KSTEPS; ++kk) {
      const int pos = kk >> 3;         // kernel position 0..8 (K-quad stays in one pos)
      const int c0 = (kk & 7) << 2;    // channel base 0..28
      const int dy = pos / 3;
      const int dx = pos - dy * 3;
      // A 16x4 f32: lanes 0-15 = {K0,K1}, lanes 16-31 = {K2,K3} (per ISA layout)
      const float2 a2 =
          *(const float2*)&imgT[((dy * TCOLS) + coli + dx) * CC + c0 + 2 * khalf];
      v2f A;
      A.x = a2.x * rn;
      A.y = a2.y * rn;
      // B 4x16 f32, pair layout kk2 = 2*kk + khalf
      const int kk2 = kk * 2 + khalf;
      const v2f B0 = *(const v2f*)&wnp[(kk2 * FF + nidx) * 2];
      const v2f B1 = *(const v2f*)&wnp[(kk2 * FF + 16 + nidx) * 2];
      acc0 = __builtin_amdgcn_wmma_f32_16x16x4_f32(false, A, false, B0, (short)0, acc0,
                                                   false, false);
      acc1 = __builtin_amdgcn_wmma_f32_16x16x4_f32(false, A, false, B1, (short)0, acc1,
                                                   false, false);
    }

    // ---- epilogue: sign(sim) * (|sim|+eps)^(p^2/100); C/D layout M = j + 8*khalf ----
    const float e0 = pexp[nidx];
    const float e1 = pexp[nidx + 16];
    const int mb = khalf * 8;
#pragma unroll
    for (int j = 0; j < 8; ++j) {
      const int wpix = wbase + mb + j;  // always < WW (tiles align to 16, WW % 16 == 0)
      const size_t o = (((size_t)b * HH + h) * WW + wpix) * FF;
      out[o + nidx] = signpow(acc0[j], e0);
      out[o + 16 + nidx] = signpow(acc1[j], e1);
    }
  }
}

extern "C" void kernel_launch(void* const* d_in, const int* in_sizes, int n_in,
                              void* d_out, int out_size, void* d_ws, size_t ws_size,
                              hipStream_t stream) {
  (void)in_sizes;
  (void)n_in;
  (void)out_size;
  (void)ws_size;
  const float* img = (const float*)d_in[0];  // (8,224,224,32) f32
  const float* w = (const float*)d_in[1];    // (1,288,32) f32
  const float* p = (const float*)d_in[2];    // (32,) f32
  const float* q = (const float*)d_in[3];    // (1,) f32
  float* out = (float*)d_out;                // (8,224,224,32) f32
  float* ws = (float*)d_ws;

  cos_prep_w<<<1, 32, 0, stream>>>(w, p, q, ws);
  dim3 grid((WW + TILE - 1) / TILE, HH, BB);  // (4, 224, 8)
  cos_sim_main<<<grid, NTHREADS, 0, stream>>>(img, ws, out);
}